// DP_66468913873579
// MI455X (gfx1250) — compile-verified
//
#include <hip/hip_runtime.h>

// PGD dual-camera update (xc, xp) for B=8, nC=28, H=256, WC=310, WP=256, STEP=2.
// Memory-bound (~465 MB @ 23.3 TB/s). Single-pass: each input byte read once,
// each output written once. Channel reductions done with V_WMMA_F32_16X16X4_F32
// via the diagonal trick (A and B operand layouts coincide for our load pattern).

typedef float v2f __attribute__((ext_vector_type(2)));
typedef float v8f __attribute__((ext_vector_type(8)));

#define NCH   28
#define HH    256
#define WCW   310
#define BB    8

// Extract diag element (p,p) of a 16x16 f32 C/D accumulator to every lane whose
// (lane&15)==p. Layout: m<8 -> VGPR m, lane m ; m>=8 -> VGPR m-8, lane m+16.
__device__ __forceinline__ float wmma_diag(const v8f acc, int p, int src) {
    float v = __shfl(acc[0], src, 32);
#pragma unroll
    for (int r = 1; r < 8; ++r) {
        float t = __shfl(acc[r], src, 32);
        v = ((p & 7) == r) ? t : v;
    }
    return v;
}

// SHIFT=false: xc update (z read directly, width W=WC).
// SHIFT=true : xp update (z read with per-channel +2c column shift from WC-wide array).
template <bool SHIFT>
__global__ void DP_psi_update(const float* __restrict__ z,     // [B,nC,H,WC] always
                              const float* __restrict__ y,     // [B,H,W]
                              const float* __restrict__ phi,   // [B,nC,H,W]
                              const float* __restrict__ prev,  // [B,nC,H,W]
                              const float* __restrict__ mu_g,
                              const float* __restrict__ muD_g,
                              float* __restrict__ out,         // [B,nC,H,W]
                              int W, int tilesPerRow) {
    const int lane   = threadIdx.x & 31;
    const int waveId = threadIdx.x >> 5;       // 8 waves / block, 16 px each
    const int half   = lane >> 4;              // 0: K={0,1}, 1: K={2,3}
    const int p      = lane & 15;              // pixel within wave tile

    const int tile = blockIdx.x % tilesPerRow;
    const int bh   = blockIdx.x / tilesPerRow;
    const int h    = bh & (HH - 1);
    const int b    = bh >> 8;

    const int w  = tile * 128 + waveId * 16 + p;
    const int wl = (w < W) ? w : (W - 1);      // clamp so EXEC stays all-ones

    const float mu  = mu_g[0];
    const float muD = muD_g[0];

    float phv[14], zv[14], pv[14];
    v8f acc_ss = {};   // sum phi^2
    v8f acc_zs = {};   // sum z * phi
    v8f acc_ps = {};   // sum prev * phi

#pragma unroll
    for (int k = 0; k < 7; ++k) {
        const int c0 = 4 * k + 2 * half;       // this lane's two channels
        const int i0 = ((b * NCH + c0) * HH + h) * W + wl;
        const int i1 = i0 + HH * W;

        const float ph0 = phi[i0];
        const float ph1 = phi[i1];
        float z0, z1;
        if (SHIFT) {
            const int zi0 = ((b * NCH + c0)     * HH + h) * WCW + wl + 2 * c0;
            const int zi1 = ((b * NCH + c0 + 1) * HH + h) * WCW + wl + 2 * (c0 + 1);
            z0 = z[zi0];  z1 = z[zi1];         // shifted-back z (w+2c <= 309 < WC)
        } else {
            z0 = z[i0];   z1 = z[i1];
        }
        const float x0 = prev[i0];
        const float x1 = prev[i1];

        phv[2 * k] = ph0; phv[2 * k + 1] = ph1;
        zv [2 * k] = z0;  zv [2 * k + 1] = z1;
        pv [2 * k] = x0;  pv [2 * k + 1] = x1;

        // Same registers serve as A (16x4, row = pixel) and B (4x16, col = pixel):
        // diag(D) accumulates the per-pixel channel dot products.
        const v2f a_phi = {ph0, ph1};
        const v2f a_z   = {z0, z1};
        const v2f a_x   = {x0, x1};
        acc_ss = __builtin_amdgcn_wmma_f32_16x16x4_f32(false, a_phi, false, a_phi,
                                                       (short)0, acc_ss, false, false);
        acc_zs = __builtin_amdgcn_wmma_f32_16x16x4_f32(false, a_z,   false, a_phi,
                                                       (short)0, acc_zs, false, false);
        acc_ps = __builtin_amdgcn_wmma_f32_16x16x4_f32(false, a_x,   false, a_phi,
                                                       (short)0, acc_ps, false, false);
    }

    const int src = (p < 8) ? p : (p + 16);
    const float ss = wmma_diag(acc_ss, p, src);   // sum phi^2
    const float sz = wmma_diag(acc_zs, p, src);   // Phi(z)
    const float sx = wmma_diag(acc_ps, p, src);   // Phi(prev)

    const float yv    = y[(b * HH + h) * W + wl];
    const float phs   = (ss == 0.0f) ? 1.0f : ss;
    const float denom = mu + muD + phs;
    const float rz    = (yv - sz) / denom;
    const float rx    = (yv - sx) / denom;
    const float wA    = muD / (mu + muD);   // weight on (z + At(rz))
    const float wB    = mu  / (mu + muD);   // weight on (prev + At(rx))

    if (w < W) {
#pragma unroll
        for (int j = 0; j < 14; ++j) {
            const int c  = 4 * (j >> 1) + 2 * half + (j & 1);
            const int oi = ((b * NCH + c) * HH + h) * W + w;
            out[oi] = wA * (zv[j] + rz * phv[j]) + wB * (pv[j] + rx * phv[j]);
        }
    }
}

extern "C" void kernel_launch(void* const* d_in, const int* in_sizes, int n_in,
                              void* d_out, int out_size, void* d_ws, size_t ws_size,
                              hipStream_t stream) {
    (void)in_sizes; (void)n_in; (void)d_ws; (void)ws_size; (void)out_size;

    const float* z      = (const float*)d_in[0];
    const float* yc     = (const float*)d_in[1];
    const float* phi_c  = (const float*)d_in[2];
    const float* yp     = (const float*)d_in[3];
    const float* phi_p  = (const float*)d_in[4];
    const float* xc_k_1 = (const float*)d_in[5];
    const float* xp_k_1 = (const float*)d_in[6];
    const float* mu     = (const float*)d_in[7];
    const float* mu_c   = (const float*)d_in[8];
    const float* mu_p   = (const float*)d_in[9];

    float* out_xc = (float*)d_out;                         // [8,28,256,310]
    float* out_xp = (float*)d_out + 8 * NCH * HH * WCW;    // [8,28,256,256]

    // xc: W=310 -> 3 tiles of 128 px per (b,h) row; 8 waves x 16 px per block.
    dim3 blk(256);
    DP_psi_update<false><<<dim3(BB * HH * 3), blk, 0, stream>>>(
        z, yc, phi_c, xp_k_1, mu, mu_c, out_xc, WCW, 3);

    // xp: W=256 -> 2 tiles of 128 px; z gathered with per-channel +2c shift.
    DP_psi_update<true><<<dim3(BB * HH * 2), blk, 0, stream>>>(
        z, yp, phi_p, xc_k_1, mu, mu_p, out_xp, 256, 2);
}